// PlanarQuantMSE_38190849196136
// MI455X (gfx1250) — compile-verified
//
#include <hip/hip_runtime.h>
#include <hip/hip_bf16.h>

#define COLS   4096
#define TPB    256
#define NCENT  16
#define NWAVES (TPB / 32)
#define CHUNKS (COLS / (TPB * 4))   // float4 chunks per thread in staging = 4

typedef __attribute__((ext_vector_type(4))) float vfloat4;
typedef __attribute__((ext_vector_type(4))) int   vint4;

__global__ __launch_bounds__(TPB) void planar_quant_kernel(
    const float* __restrict__ x,
    const float* __restrict__ centroids,
    const float* __restrict__ rot2,
    float* __restrict__ xhat,
    int* __restrict__ indices)
{
    __shared__ float rowbuf[COLS];     // 16 KB staged row
    __shared__ float centLds[NCENT];   // 16 centroids for per-lane gather
    __shared__ float redbuf[NWAVES];   // cross-wave reduction

    const int    tid  = threadIdx.x;
    const size_t row  = blockIdx.x;
    const float* xrow = x + row * (size_t)COLS;

    if (tid < NCENT) centLds[tid] = centroids[tid];

    // ---- Stage 16 KB row into LDS via CDNA5 async global->LDS DMA ----
#if defined(__HIP_DEVICE_COMPILE__) && defined(__gfx1250__)
    #pragma unroll
    for (int i = 0; i < CHUNKS; ++i) {
        const int    c4 = tid + i * TPB;               // float4 chunk index
        const float* gp = xrow + (size_t)c4 * 4;
        unsigned     lp = (unsigned)(size_t)(&rowbuf[c4 * 4]); // LDS byte addr
        asm volatile("global_load_async_to_lds_b128 %0, %1, off"
                     :: "v"(lp), "v"(gp) : "memory");
    }
    asm volatile("s_wait_asynccnt 0" ::: "memory");
#else
    #pragma unroll
    for (int i = 0; i < CHUNKS; ++i) {
        const int c4 = tid + i * TPB;
        ((vfloat4*)rowbuf)[c4] = __builtin_nontemporal_load((const vfloat4*)xrow + c4);
    }
#endif
    __syncthreads();

    // ---- Pull my 16 contiguous elements from LDS into registers ----
    float v[16];
    #pragma unroll
    for (int k = 0; k < 4; ++k) {
        vfloat4 t = ((const vfloat4*)rowbuf)[tid * 4 + k];
        v[4 * k + 0] = t.x; v[4 * k + 1] = t.y;
        v[4 * k + 2] = t.z; v[4 * k + 3] = t.w;
    }

    // ---- Row sum-of-squares: per-thread, wave32 shuffle, then cross-wave ----
    float ss = 0.0f;
    #pragma unroll
    for (int j = 0; j < 16; ++j) ss = fmaf(v[j], v[j], ss);
    #pragma unroll
    for (int off = 16; off >= 1; off >>= 1)
        ss += __shfl_xor(ss, off, 32);
    if ((tid & 31) == 0) redbuf[tid >> 5] = ss;
    __syncthreads();

    float tot = 0.0f;
    #pragma unroll
    for (int w = 0; w < NWAVES; ++w) tot += redbuf[w];
    const float norm = fmaxf(sqrtf(tot), 1e-8f);

    // ---- Midpoints of sorted centroids, pre-scaled by the row norm.
    //      count(rot(v/norm) > mid)  ==  count(rot(v) > mid*norm)   (norm > 0)
    //      -> kills the divide and the per-element normalize multiplies.
    float midn[NCENT - 1];
    {
        float c[NCENT];
        #pragma unroll
        for (int k = 0; k < 4; ++k) {
            vfloat4 t = ((const vfloat4*)centLds)[k];
            c[4 * k + 0] = t.x; c[4 * k + 1] = t.y;
            c[4 * k + 2] = t.z; c[4 * k + 3] = t.w;
        }
        #pragma unroll
        for (int m = 0; m < NCENT - 1; ++m)
            midn[m] = (0.5f * (c[m] + c[m + 1])) * norm;
    }

    float* xo = xhat    + row * (size_t)COLS;
    int*   io = indices + row * (size_t)COLS;
    const vfloat4* rot4 = (const vfloat4*)rot2;   // (c0,s0,c1,s1) per 16B, L2-hot

    #pragma unroll
    for (int k = 0; k < 4; ++k) {
        vfloat4 cs = rot4[tid * 4 + k];
        float o[4]; int qi[4];
        #pragma unroll
        for (int p = 0; p < 2; ++p) {
            const float cc = p ? cs.z : cs.x;
            const float sn = p ? cs.w : cs.y;
            const float v0 = v[4 * k + 2 * p + 0];
            const float v1 = v[4 * k + 2 * p + 1];
            const float r0 = fmaf(cc, v0, -sn * v1);   //  norm * ( c*u0 - s*u1)
            const float r1 = fmaf(sn, v0,  cc * v1);   //  norm * ( s*u0 + c*u1)
            int i0 = 0, i1 = 0;
            #pragma unroll
            for (int m = 0; m < NCENT - 1; ++m) {      // strict '>' => first-index ties
                i0 += (r0 > midn[m]) ? 1 : 0;
                i1 += (r1 > midn[m]) ? 1 : 0;
            }
            const float q0 = centLds[i0];              // ds_load_b32 gather
            const float q1 = centLds[i1];
            const float f0 = fmaf( cc, q0, sn * q1);   //  c*q0 + s*q1
            const float f1 = fmaf(-sn, q0, cc * q1);   // -s*q0 + c*q1
            o[2 * p + 0] = f0 * norm;
            o[2 * p + 1] = f1 * norm;
            qi[2 * p + 0] = i0;
            qi[2 * p + 1] = i1;
        }
        vfloat4 ov = { o[0], o[1], o[2], o[3] };
        vint4   iv = { qi[0], qi[1], qi[2], qi[3] };
        __builtin_nontemporal_store(ov, (vfloat4*)xo + tid * 4 + k);
        __builtin_nontemporal_store(iv, (vint4*)io + tid * 4 + k);
    }
}

extern "C" void kernel_launch(void* const* d_in, const int* in_sizes, int n_in,
                              void* d_out, int out_size, void* d_ws, size_t ws_size,
                              hipStream_t stream) {
    const float* x         = (const float*)d_in[0];
    const float* centroids = (const float*)d_in[1];
    const float* rot2      = (const float*)d_in[2];

    const int d    = in_sizes[2];           // 2 * n_groups = 4096 columns
    const int rows = in_sizes[0] / d;       // 4096 rows

    float* xhat    = (float*)d_out;
    int*   indices = (int*)((float*)d_out + (size_t)rows * (size_t)d);

    planar_quant_kernel<<<rows, TPB, 0, stream>>>(x, centroids, rot2, xhat, indices);
}